// ChamferLoss_26877905338419
// MI455X (gfx1250) — compile-verified
//
#include <hip/hip_runtime.h>
#include <math.h>

// Chamfer loss, B=16, K=4096, D=3, via V_WMMA_F32_16X16X4_F32.
//
// min_j d2(i,j) = x2_i + min_j (y2_j - 2 x_i.y_j).
// The bracket is ONE WMMA with C = inline 0 by folding everything into the
// padded K=4 slot:   A row i (K=0..3) = (-2x0, -2x1, -2x2, 1.0)
//                    B col j (K=0..3) = ( y0,   y1,   y2,  y2norm+pen)
// => D = -2 x.y + y2 + colmask_sentinel.  x2_i is added once at store time
// (row mask only gates the final sum, never the min).

#define BB 16
#define KK 4096
#define BK (BB * KK)                       // 65536 points total
#define DTILES_PER_BATCH (KK / 32)         // 128 double-row-tiles per batch
#define WAVES_PER_BLOCK 8
#define TOTAL_WAVES (BB * DTILES_PER_BATCH)          // 2048
#define NBLOCKS_MAIN (TOTAL_WAVES / WAVES_PER_BLOCK) // 256

typedef __attribute__((ext_vector_type(2))) float v2f;
typedef __attribute__((ext_vector_type(8))) float v8f;

// xor-lane min via ds_swizzle (group-of-32: and=0x1f, or=0, xor=m).
#define XORMIN(v, m)                                                       \
  fminf((v), __int_as_float(__builtin_amdgcn_ds_swizzle(                   \
                 __float_as_int(v), (((m) << 10) | 0x1f))))

// Per point: pure norm+sentinel (for the row-side add) and the B operand
// packed in WMMA per-lane layout:
//   pack[0*BK + idx] = (c0, c1)            (lanes 0-15:  K=0,1)
//   pack[1*BK + idx] = (c2, norm+pen)      (lanes 16-31: K=2,3)
__global__ __launch_bounds__(256) void chamfer_prep_kernel(
    const float* __restrict__ P, const float* __restrict__ T,
    const unsigned char* __restrict__ mask,
    float* __restrict__ normP, float* __restrict__ normT,
    v2f* __restrict__ packP, v2f* __restrict__ packT)
{
  const int idx = blockIdx.x * 256 + threadIdx.x;   // < BK exactly
  const float pen = mask[idx] ? 0.0f : 1e30f;

  const float p0 = P[3 * idx], p1 = P[3 * idx + 1], p2 = P[3 * idx + 2];
  const float np = fmaf(p0, p0, fmaf(p1, p1, p2 * p2)) + pen;
  normP[idx]      = np;
  packP[idx]      = (v2f){p0, p1};
  packP[BK + idx] = (v2f){p2, np};

  const float t0 = T[3 * idx], t1 = T[3 * idx + 1], t2 = T[3 * idx + 2];
  const float nt = fmaf(t0, t0, fmaf(t1, t1, t2 * t2)) + pen;
  normT[idx]      = nt;
  packT[idx]      = (v2f){t0, t1};
  packT[BK + idx] = (v2f){t2, nt};
}

// One wave owns TWO 16-row tiles of X (32 rows) and sweeps all 256 column
// tiles of Y; each b64 B-tile load feeds two WMMAs with C = inline 0.
__global__ __launch_bounds__(256) void chamfer_dir_kernel(
    const float* __restrict__ X, const float* __restrict__ xnorm,
    const v2f* __restrict__ ypack, float* __restrict__ minout)
{
  const int tid  = threadIdx.x;
  const int lane = tid & 31;
  const int g    = lane >> 4;   // half-wave: A/B hold K=0,1 vs K=2,3
  const int n    = lane & 15;   // row idx (A) / col idx (B,C,D) within tile
  const int wave = blockIdx.x * WAVES_PER_BLOCK + (tid >> 5);
  const int b    = wave >> 7;                       // / DTILES_PER_BATCH
  const int t    = wave & (DTILES_PER_BATCH - 1);

  const int row0 = b * KK + t * 32 + n;   // first tile's rows
  const int row1 = row0 + 16;             // second tile's rows

  // A tiles (16x4 f32): K=0..2 = -2*x, K=3 = 1.0 (picks up y2+pen from B).
  v2f a0, a1;
  {
    const float x0 = X[row0 * 3], x1 = X[row0 * 3 + 1], x2 = X[row0 * 3 + 2];
    a0.x = (g ? x2 : x0) * -2.0f;
    a0.y = g ? 1.0f : (x1 * -2.0f);
    const float z0 = X[row1 * 3], z1 = X[row1 * 3 + 1], z2 = X[row1 * 3 + 2];
    a1.x = (g ? z2 : z0) * -2.0f;
    a1.y = g ? 1.0f : (z1 * -2.0f);
  }

  float rm0[8], rm1[8];
#pragma unroll
  for (int r = 0; r < 8; ++r) { rm0[r] = 3.0e38f; rm1[r] = 3.0e38f; }

  const v2f* yp = ypack + (size_t)g * BK + b * KK;  // this half-wave's packing
  const v8f zc = {};                                // C = 0 (inline)

  for (int J = 0; J < KK / 16; ++J) {
    const v2f bm = yp[J * 16 + n];   // one b64 load per tile column set

    v8f c0 = __builtin_amdgcn_wmma_f32_16x16x4_f32(false, a0, false, bm,
                                                   (short)0, zc, false, false);
    v8f c1 = __builtin_amdgcn_wmma_f32_16x16x4_f32(false, a1, false, bm,
                                                   (short)0, zc, false, false);
#pragma unroll
    for (int r = 0; r < 8; ++r) {
      rm0[r] = fminf(rm0[r], c0[r]);
      rm1[r] = fminf(rm1[r], c1[r]);
    }
  }

  // Min across the 16 column-classes in each half-wave (xor swizzles stay
  // within a 16-lane half); add x2[row] at store. One writer per row.
  const int base = b * KK + t * 32;
#pragma unroll
  for (int r = 0; r < 8; ++r) {
    float v = rm0[r];
    v = XORMIN(v, 1); v = XORMIN(v, 2); v = XORMIN(v, 4); v = XORMIN(v, 8);
    float w = rm1[r];
    w = XORMIN(w, 1); w = XORMIN(w, 2); w = XORMIN(w, 4); w = XORMIN(w, 8);
    if (n == r) {
      const int m = r + 8 * g;
      minout[base + m]      = v + xnorm[base + m];
      minout[base + 16 + m] = w + xnorm[base + 16 + m];
    }
  }
}

// Final masked sum of sqrt(min d2) for both directions + valid count.
__global__ __launch_bounds__(256) void chamfer_reduce_kernel(
    const float* __restrict__ minP, const float* __restrict__ minT,
    const unsigned char* __restrict__ mask, float* __restrict__ out)
{
  __shared__ float ssum[256];
  __shared__ float scnt[256];
  float s = 0.0f, cnt = 0.0f;
  for (int idx = threadIdx.x; idx < BK; idx += 256) {
    if (mask[idx]) {
      cnt += 1.0f;
      s += sqrtf(fmaxf(minP[idx], 1e-12f)) + sqrtf(fmaxf(minT[idx], 1e-12f));
    }
  }
  ssum[threadIdx.x] = s;
  scnt[threadIdx.x] = cnt;
  __syncthreads();
  for (int o = 128; o > 0; o >>= 1) {
    if (threadIdx.x < o) {
      ssum[threadIdx.x] += ssum[threadIdx.x + o];
      scnt[threadIdx.x] += scnt[threadIdx.x + o];
    }
    __syncthreads();
  }
  if (threadIdx.x == 0)
    out[0] = ssum[0] / (2.0f * (scnt[0] + 1e-8f));
}

extern "C" void kernel_launch(void* const* d_in, const int* in_sizes, int n_in,
                              void* d_out, int out_size, void* d_ws, size_t ws_size,
                              hipStream_t stream) {
  (void)in_sizes; (void)n_in; (void)out_size; (void)ws_size;
  const float* pred         = (const float*)d_in[0];
  const float* target       = (const float*)d_in[1];
  const unsigned char* mask = (const unsigned char*)d_in[2];
  float* out = (float*)d_out;

  // Workspace layout (floats): normP[BK] | normT[BK] | packP[2*BK v2f]
  //                            | packT[2*BK v2f] | minP[BK] | minT[BK]  (3 MB)
  float* normP = (float*)d_ws;
  float* normT = normP + BK;
  v2f*   packP = (v2f*)(normT + BK);
  v2f*   packT = packP + 2 * BK;
  float* minP  = (float*)(packT + 2 * BK);
  float* minT  = minP + BK;

  chamfer_prep_kernel<<<BK / 256, 256, 0, stream>>>(pred, target, mask,
                                                    normP, normT, packP, packT);
  // p -> t: rows from pred, columns from target
  chamfer_dir_kernel<<<NBLOCKS_MAIN, 256, 0, stream>>>(pred, normP, packT, minP);
  // t -> p: rows from target, columns from pred
  chamfer_dir_kernel<<<NBLOCKS_MAIN, 256, 0, stream>>>(target, normT, packP, minT);
  chamfer_reduce_kernel<<<1, 256, 0, stream>>>(minP, minT, mask, out);
}